// ModelnetClassifier_58445914964597
// MI455X (gfx1250) — compile-verified
//
#include <hip/hip_runtime.h>

#if defined(__HIP_DEVICE_COMPILE__) && __has_builtin(__builtin_amdgcn_tensor_load_to_lds)
#define USE_TDM 1
#else
#define USE_TDM 0
#endif

namespace {

constexpr int Bv = 4;     // batch
constexpr int Nv = 4096;  // vertices
constexpr int Rr = 5;     // radial bins
constexpr int Aa = 8;     // angular bins
constexpr int Kk = 3;     // barycentric verts
constexpr int TILE_M = 16;

typedef __attribute__((ext_vector_type(2))) float v2f;
typedef __attribute__((ext_vector_type(8))) float v8f;

#if USE_TDM
typedef __attribute__((ext_vector_type(4))) unsigned int u32x4;
typedef __attribute__((ext_vector_type(4))) int i32x4;
typedef __attribute__((ext_vector_type(8))) int i32x8;

// TDM: DMA a contiguous run of nElems fp32 from global into LDS.
// D# group0: count=1, lds_addr, 57-bit global addr, type=2 ("image").
// D# group1: data_size=4B, tensor_dim0 = tile_dim0 = nElems, tensor_dim1 = tile_dim1 = 1.
__device__ __forceinline__ void tdm_load_1d(const float* gsrc, const float* lds_dst,
                                            int nElems) {
  unsigned long long ga = (unsigned long long)gsrc;
  unsigned lofs = (unsigned)(unsigned long long)
      (const __attribute__((address_space(3))) float*)lds_dst;  // raw LDS byte offset
  u32x4 g0;
  g0.x = 1u;                                                    // count = 1 (valid)
  g0.y = lofs;                                                  // lds_addr
  g0.z = (unsigned)(ga & 0xffffffffull);                        // global_addr[31:0]
  g0.w = ((unsigned)((ga >> 32) & 0x1ffffffull)) | (2u << 30);  // addr[56:32] | type=2
  i32x8 g1;
  g1[0] = 0x00020000;                                   // data_size = 2 (4 bytes)
  g1[1] = (int)(((unsigned)nElems & 0xffffu) << 16);    // tensor_dim0[15:0]  @bit48
  g1[2] = (int)(((unsigned)nElems >> 16) | (1u << 16)); // dim0[31:16] | tensor_dim1=1
  g1[3] = (int)(((unsigned)nElems & 0xffffu) << 16);    // tile_dim0 @bit112
  g1[4] = 1;                                            // tile_dim1 = 1
  g1[5] = nElems;                                       // tensor_dim0_stride
  g1[6] = 0;
  g1[7] = 0;
  i32x4 gz = {0, 0, 0, 0};
#if __clang_major__ >= 23
  i32x8 gz8 = {0, 0, 0, 0, 0, 0, 0, 0};
  __builtin_amdgcn_tensor_load_to_lds(g0, g1, gz, gz, gz8, 0);
#else
  __builtin_amdgcn_tensor_load_to_lds(g0, g1, gz, gz, 0);
#endif
}
#endif  // USE_TDM

// One workgroup = 8 waves = 256 threads; owns TILE_M=16 vertices of one batch.
// Wave w accumulates rotation-shift s=w with fp32 WMMA; the W[r][aw] tile is
// shared by all 8 waves and staged into LDS (TDM double-buffered).
template <int C, int D>
__global__ __launch_bounds__(256) void isc_layer_kernel(
    const float* __restrict__ x,       // [B,N,C]
    const int*   __restrict__ bc_idx,  // [B,N,R,A,K]
    const float* __restrict__ bc_w,    // [B,N,R,A,K]
    const float* __restrict__ W,       // [R,A,C,D]
    const float* __restrict__ bias,    // [D]
    float*       __restrict__ y_out)   // [B,N,D]
{
  constexpr int CPAD = (C + 3) & ~3;        // K-dim padded to multiple of 4
  constexpr int NT   = (D + 15) / 16;       // 16-wide output tiles per shift
  constexpr bool DP  = (D % 16) != 0;
  constexpr int SHAT_F = Aa * CPAD * TILE_M;   // shat stored [a][c][m] (bank-friendly)
  constexpr int WT_F   = CPAD * D;             // one staged weight tile [c][d]
  constexpr int YRED_F = Aa * TILE_M * D;      // per-shift outputs for max-pool
  constexpr int MAIN_F = SHAT_F + 2 * WT_F;
  constexpr int LDS_F  = MAIN_F > YRED_F ? MAIN_F : YRED_F;
  __shared__ float lds[LDS_F];
  float* shat = lds;  // scalar LDS-derived pointers only (no aggregates!)

  const int tid   = threadIdx.x;
  const int wave  = tid >> 5;   // rotation shift s owned by this wave
  const int lane  = tid & 31;
  const int m16   = lane & 15;  // matrix row (A) / col (B,D)
  const int khalf = lane >> 4;  // K-half of the 16x4 / 4x16 fragment

  const int bb = blockIdx.x / (Nv / TILE_M);
  const int n0 = (blockIdx.x % (Nv / TILE_M)) * TILE_M;

  v8f acc[NT] = {};

  // phase-1 split: 128 (m,a) pairs (m fastest -> conflict-free LDS writes), 2 halves
  const int pair  = tid & 127;
  const int chalf = tid >> 7;
  const int pm = pair & 15;
  const int pa = pair >> 4;

  for (int r = 0; r < Rr; ++r) {
#if USE_TDM
    if (wave == 0)  // kick off W[r][0] DMA; overlaps the gather phase below
      tdm_load_1d(W + (size_t)(r * Aa) * C * D, lds + SHAT_F, C * D);
#endif
    // ---- phase 1: gather + barycentric interpolation -> shat[a][c][m] ----
    {
      const int base = ((((bb * Nv) + n0 + pm) * Rr + r) * Aa + pa) * Kk;
      const int i0 = bc_idx[base + 0];
      const int i1 = bc_idx[base + 1];
      const int i2 = bc_idx[base + 2];
      const float w0 = bc_w[base + 0];
      const float w1 = bc_w[base + 1];
      const float w2 = bc_w[base + 2];
      const float* x0 = x + (bb * Nv + i0) * C;
      const float* x1 = x + (bb * Nv + i1) * C;
      const float* x2 = x + (bb * Nv + i2) * C;
      const int c0 = chalf * (CPAD / 2);
      #pragma unroll 4
      for (int c = c0; c < c0 + CPAD / 2; ++c) {
        float v = 0.0f;
        if (C % 4 == 0 || c < C)  // zero-pad channels to CPAD
          v = x0[c] * w0 + x1[c] * w1 + x2[c] * w2;
        shat[(pa * CPAD + c) * TILE_M + pm] = v;
      }
    }
    __syncthreads();  // shat ready (previous r's readers passed their exit barrier)

    for (int aw = 0; aw < Aa; ++aw) {
      const float* wb = lds + SHAT_F + (aw & 1) * WT_F;  // current staged W tile
#if USE_TDM
      if (wave == 0) {
        if (aw + 1 < Aa) {  // prefetch next tile into the other buffer, wait current
          tdm_load_1d(W + (size_t)(r * Aa + aw + 1) * C * D,
                      lds + SHAT_F + ((aw + 1) & 1) * WT_F, C * D);
          __builtin_amdgcn_s_wait_tensorcnt(1);
        } else {
          __builtin_amdgcn_s_wait_tensorcnt(0);
        }
      }
#else
      {  // cooperative staging fallback
        const float* Wr = W + (size_t)(r * Aa + aw) * C * D;
        float* wdst = lds + SHAT_F + (aw & 1) * WT_F;
        for (int e = tid; e < C * D; e += 256) wdst[e] = Wr[e];
      }
#endif
      __syncthreads();  // W tile visible to all waves

      // ---- phase 2: rotational correlation, fp32 WMMA 16x16x4 ----
      const int asrc = (aw + wave) & 7;  // roll(shat, -s) along angular axis
      #pragma unroll 4
      for (int c4 = 0; c4 < CPAD; c4 += 4) {
        const int cA = c4 + 2 * khalf;
        v2f afrag;  // 16 lanes -> 16 consecutive dwords: conflict-free
        afrag.x = shat[(asrc * CPAD + cA) * TILE_M + m16];
        afrag.y = shat[(asrc * CPAD + cA + 1) * TILE_M + m16];
        const float mcx = (C % 4 == 0 || cA < C) ? 1.0f : 0.0f;      // branchless pad
        const float mcy = (C % 4 == 0 || cA + 1 < C) ? 1.0f : 0.0f;
        #pragma unroll
        for (int nt = 0; nt < NT; ++nt) {
          const int dcol = nt * 16 + m16;
          const int didx = DP ? (dcol & (D - 1)) : dcol;             // clamped index
          const float md = (!DP || dcol < D) ? 1.0f : 0.0f;
          v2f bfrag;
          bfrag.x = wb[cA * D + didx] * (mcx * md);
          bfrag.y = wb[(cA + 1) * D + didx] * (mcy * md);
          acc[nt] = __builtin_amdgcn_wmma_f32_16x16x4_f32(
              false, afrag, false, bfrag, (short)0, acc[nt], false, false);
        }
      }
      __syncthreads();  // protect wbuf[aw&1] before tile aw+2 overwrites it
    }
  }

  // ---- phase 3: cross-shift max pool + bias + relu (reuse LDS) ----
  #pragma unroll
  for (int nt = 0; nt < NT; ++nt) {
    #pragma unroll
    for (int v = 0; v < 8; ++v) {
      const int mrow = v + 8 * khalf;  // D-matrix VGPR layout
      const int dcol = nt * 16 + m16;
      if (!DP || dcol < D)
        lds[(wave * TILE_M + mrow) * D + dcol] = acc[nt][v];
    }
  }
  __syncthreads();
  for (int e = tid; e < TILE_M * D; e += 256) {
    const int m = e / D, d = e % D;
    float mx = lds[m * D + d];
    #pragma unroll
    for (int s = 1; s < Aa; ++s)
      mx = fmaxf(mx, lds[(s * TILE_M + m) * D + d]);
    y_out[((bb * Nv) + n0 + m) * D + d] = fmaxf(mx + bias[d], 0.0f);
  }
}

// Per-batch covariance of x3 [B,N,8] + dense head [64,40]. One WG per batch.
__global__ __launch_bounds__(256) void cov_dense_kernel(
    const float* __restrict__ x,    // [B,N,8]
    const float* __restrict__ Wd,   // [64,40]
    const float* __restrict__ bd,   // [40]
    float*       __restrict__ out)  // [B,40]
{
  const int b = blockIdx.x, tid = threadIdx.x;
  float s[8];
  float p[8][8];
  #pragma unroll
  for (int c = 0; c < 8; ++c) {
    s[c] = 0.0f;
    #pragma unroll
    for (int d = 0; d < 8; ++d) p[c][d] = 0.0f;
  }
  for (int n = tid; n < Nv; n += 256) {
    const float* xp = x + ((size_t)b * Nv + n) * 8;
    float v[8];
    #pragma unroll
    for (int c = 0; c < 8; ++c) { v[c] = xp[c]; s[c] += v[c]; }
    #pragma unroll
    for (int c = 0; c < 8; ++c)
      #pragma unroll
      for (int d = 0; d < 8; ++d) p[c][d] += v[c] * v[d];
  }
  __shared__ float red[8 + 64];
  if (tid < 72) red[tid] = 0.0f;
  __syncthreads();
  #pragma unroll
  for (int c = 0; c < 8; ++c) atomicAdd(&red[c], s[c]);
  #pragma unroll
  for (int c = 0; c < 8; ++c)
    #pragma unroll
    for (int d = 0; d < 8; ++d) atomicAdd(&red[8 + c * 8 + d], p[c][d]);
  __syncthreads();
  if (tid < 40) {
    float mean[8];
    #pragma unroll
    for (int c = 0; c < 8; ++c) mean[c] = red[c] * (1.0f / Nv);
    float acc = bd[tid];
    #pragma unroll
    for (int c = 0; c < 8; ++c)
      #pragma unroll
      for (int d = 0; d < 8; ++d) {
        const float cv = red[8 + c * 8 + d] * (1.0f / Nv) - mean[c] * mean[d];
        acc += cv * Wd[(c * 8 + d) * 40 + tid];
      }
    out[b * 40 + tid] = acc;
  }
}

}  // namespace

extern "C" void kernel_launch(void* const* d_in, const int* in_sizes, int n_in,
                              void* d_out, int out_size, void* d_ws, size_t ws_size,
                              hipStream_t stream) {
  (void)in_sizes; (void)n_in; (void)out_size; (void)ws_size;
  const float* signal = (const float*)d_in[0];   // [B,N,3]
  const int*   bc_idx = (const int*)  d_in[1];   // [B,N,R,A,K]
  const float* bc_w   = (const float*)d_in[2];   // [B,N,R,A,K]
  const float* W1 = (const float*)d_in[3];
  const float* b1 = (const float*)d_in[4];
  const float* W2 = (const float*)d_in[5];
  const float* b2 = (const float*)d_in[6];
  const float* W3 = (const float*)d_in[7];
  const float* b3 = (const float*)d_in[8];
  const float* Wd = (const float*)d_in[9];
  const float* bd = (const float*)d_in[10];

  float* x1 = (float*)d_ws;                // [B,N,128]  8 MB
  float* x2 = x1 + (size_t)Bv * Nv * 128;  // [B,N,64]   4 MB
  float* x3 = x2 + (size_t)Bv * Nv * 64;   // [B,N,8]    0.5 MB

  const dim3 grid(Bv * (Nv / TILE_M));  // 1024 workgroups
  isc_layer_kernel<3, 128><<<grid, 256, 0, stream>>>(signal, bc_idx, bc_w, W1, b1, x1);
  isc_layer_kernel<128, 64><<<grid, 256, 0, stream>>>(x1, bc_idx, bc_w, W2, b2, x2);
  isc_layer_kernel<64, 8><<<grid, 256, 0, stream>>>(x2, bc_idx, bc_w, W3, b3, x3);
  cov_dense_kernel<<<Bv, 256, 0, stream>>>(x3, Wd, bd, (float*)d_out);
}